// SAGE_encoder_24438363914372
// MI455X (gfx1250) — compile-verified
//
#include <hip/hip_runtime.h>
#include <hip/hip_bf16.h>

typedef __attribute__((ext_vector_type(2))) float v2f;
typedef __attribute__((ext_vector_type(8))) float v8f;

#define F_DIM 128
#define H_DIM 128

// ds_swizzle_b32 group-of-32 XOR exchange: pattern = (xor_mask<<10) | and_mask
template <int XORMASK>
__device__ __forceinline__ float swz_xor(float v) {
    int r = __builtin_amdgcn_ds_swizzle(__float_as_int(v), (XORMASK << 10) | 0x1F);
    return __int_as_float(r);
}

// ---------------------------------------------------------------------------
// k_zero: clear workspace (agg, deg, colsum, colsumsq)
// ---------------------------------------------------------------------------
__global__ void k_zero(float* __restrict__ ws, long total) {
    long i = (long)blockIdx.x * blockDim.x + threadIdx.x;
    if (i < total) ws[i] = 0.0f;
}

// ---------------------------------------------------------------------------
// k_pack: repack [W_l ; W_r] (256x128 K-major concat) into WMMA B-fragment
// order: Wt[kk][half][n] = { W[4kk+2*half][n], W[4kk+2*half+1][n] }  (float2)
// so lane L of a wave loads its B frag as ONE contiguous b64 at a constant
// 2048-byte stride per K-step. 64*2*128 float2 = 128 KB, L2-resident.
// ---------------------------------------------------------------------------
__global__ void k_pack(const float* __restrict__ Wl,
                       const float* __restrict__ Wr,
                       float2* __restrict__ Wt) {
    int i = blockIdx.x * blockDim.x + threadIdx.x;   // 0 .. 16383
    if (i >= 64 * 2 * 128) return;
    int n    = i & 127;
    int half = (i >> 7) & 1;
    int kk   = i >> 8;
    int k    = kk * 4 + half * 2;                    // even; k and k+1 same matrix
    const float* W = (k < 128) ? Wl : Wr;
    int kr = k & 127;
    Wt[i] = make_float2(W[kr * H_DIM + n], W[(kr + 1) * H_DIM + n]);
}

// ---------------------------------------------------------------------------
// k_scatter: one wave (32 lanes) per edge. Each lane handles 4 consecutive
// features (float4 gather of x[src], 4 global f32 atomics into agg[dst]).
// Lane 0 also counts the edge into deg[dst].
// ---------------------------------------------------------------------------
__global__ void k_scatter(const float* __restrict__ x,
                          const int*   __restrict__ ei,
                          float* __restrict__ agg,
                          float* __restrict__ deg,
                          int E) {
    long t = (long)blockIdx.x * blockDim.x + threadIdx.x;
    int e    = (int)(t >> 5);
    int lane = (int)(t & 31);
    if (e >= E) return;
    int src = ei[e];
    int dst = ei[(long)E + e];
    float4 v = ((const float4*)(x + (long)src * F_DIM))[lane];
    float* a = agg + (long)dst * F_DIM + lane * 4;
    atomicAdd(a + 0, v.x);
    atomicAdd(a + 1, v.y);
    atomicAdd(a + 2, v.z);
    atomicAdd(a + 3, v.w);
    if (lane == 0) atomicAdd(deg + dst, 1.0f);
}

// ---------------------------------------------------------------------------
// k_gemm: block = 256 threads = 8 wave32; block computes 16 full output rows.
//   out[16,128] = [agg/deg | x] (16x256) @ [W_l ; W_r] (256x128) + b_l
// then row-wise L2 normalize, ReLU, store, and accumulate per-column
// sum / sum-of-squares for BatchNorm.
// Wave w computes the 16x16 tile at columns [16w,16w+16): 64x
// V_WMMA_F32_16X16X4_F32, A frags from LDS (ds_load_2addr_b64 pairs),
// B frags as single b64 loads from the repacked Wt.
// Requires N % 16 == 0 (true: 100000/16 = 6250).
// ---------------------------------------------------------------------------
__global__ __launch_bounds__(256)
void k_gemm(const float* __restrict__ x,
            const float* __restrict__ agg,
            const float* __restrict__ deg,
            const v2f*   __restrict__ Wt,
            const float* __restrict__ bl,
            float* __restrict__ out,
            float* __restrict__ colsum,
            float* __restrict__ colsumsq) {
    __shared__ float A[16 * 256];     // [agg/deg | x] tile, row stride 256
    __shared__ float invdeg[16];
    __shared__ float nrm[16];
    __shared__ float invnrm[16];

    const int tid  = threadIdx.x;
    const int row0 = blockIdx.x * 16;

    if (tid < 16) {
        float d = deg[row0 + tid];
        invdeg[tid] = 1.0f / fmaxf(d, 1.0f);
        nrm[tid] = 0.0f;
    }
    __syncthreads();

    // Stage A tile: cols 0..127 = mean-aggregated neighbors, 128..255 = x
    for (int i = tid; i < 16 * 128; i += 256) {
        int m = i >> 7;
        int f = i & 127;
        long r = (long)(row0 + m) * F_DIM + f;
        A[m * 256 + f]       = agg[r] * invdeg[m];
        A[m * 256 + 128 + f] = x[r];
    }
    __syncthreads();

    const int wave = tid >> 5;
    const int lane = tid & 31;
    const int half = lane >> 4;   // 0: K{0,1}/C rows 0-7, 1: K{2,3}/C rows 8-15
    const int col  = lane & 15;
    const int n    = wave * 16 + col;
    const int koff = half * 2;

    v8f c = {0.f, 0.f, 0.f, 0.f, 0.f, 0.f, 0.f, 0.f};
    const float* Abase = &A[col * 256 + koff];
    const v2f*   Bbase = Wt + half * 128 + n;     // stride 256 v2f per K-step

#pragma unroll 8
    for (int kk = 0; kk < 64; ++kk) {
        v2f a = *(const v2f*)(Abase + 4 * kk);
        v2f b = Bbase[kk * 256];
        c = __builtin_amdgcn_wmma_f32_16x16x4_f32(false, a, false, b,
                                                  (short)0, c, false, false);
    }

    // Epilogue: bias, then row L2-norm over all 128 columns.
    // C layout: VGPR r, lanes 0-15 -> (M=r, N=col), lanes 16-31 -> (M=r+8, N=col)
    // XOR masks 1/2/4/8 stay inside each 16-lane half; ds_swizzle does the
    // exchange in one instruction per step (EXEC is all-1s here).
    const float bias = bl[n];
    float vals[8];
#pragma unroll
    for (int r = 0; r < 8; ++r) {
        vals[r] = c[r] + bias;
        float s = vals[r] * vals[r];
        s += swz_xor<1>(s);
        s += swz_xor<2>(s);
        s += swz_xor<4>(s);
        s += swz_xor<8>(s);
        if (col == 0) atomicAdd(&nrm[r + 8 * half], s);
    }
    __syncthreads();
    if (tid < 16) invnrm[tid] = 1.0f / fmaxf(sqrtf(nrm[tid]), 1e-12f);
    __syncthreads();

    float csum = 0.0f, csq = 0.0f;
#pragma unroll
    for (int r = 0; r < 8; ++r) {
        int m = r + 8 * half;
        float v = vals[r] * invnrm[m];
        v = fmaxf(v, 0.0f);                       // ReLU after normalize
        out[(long)(row0 + m) * H_DIM + n] = v;
        csum += v;
        csq  += v * v;
    }
    // lanes L and L+16 hold the same column n -> fold halves, then one atomic
    csum += swz_xor<16>(csum);
    csq  += swz_xor<16>(csq);
    if (half == 0) {
        atomicAdd(&colsum[n], csum);
        atomicAdd(&colsumsq[n], csq);
    }
}

// ---------------------------------------------------------------------------
// k_bn: finalize BatchNorm in place (float4 per thread, columns n..n+3)
// ---------------------------------------------------------------------------
__global__ void k_bn(float* __restrict__ out,
                     const float* __restrict__ colsum,
                     const float* __restrict__ colsumsq,
                     const float* __restrict__ gamma,
                     const float* __restrict__ beta,
                     int N) {
    long t = (long)blockIdx.x * blockDim.x + threadIdx.x;
    long total = (long)N * (H_DIM / 4);
    if (t >= total) return;
    long base = t * 4;
    int n = (int)(base & (H_DIM - 1));
    float invN = 1.0f / (float)N;
    float4 v = *(float4*)(out + base);
    float o[4] = {v.x, v.y, v.z, v.w};
#pragma unroll
    for (int j = 0; j < 4; ++j) {
        float mean = colsum[n + j] * invN;
        float var  = colsumsq[n + j] * invN - mean * mean;
        float inv  = rsqrtf(var + 1e-5f);
        o[j] = (o[j] - mean) * inv * gamma[n + j] + beta[n + j];
    }
    *(float4*)(out + base) = make_float4(o[0], o[1], o[2], o[3]);
}

// ---------------------------------------------------------------------------
extern "C" void kernel_launch(void* const* d_in, const int* in_sizes, int n_in,
                              void* d_out, int out_size, void* d_ws, size_t ws_size,
                              hipStream_t stream) {
    const float* x     = (const float*)d_in[0];
    const int*   ei    = (const int*)d_in[1];
    const float* Wl    = (const float*)d_in[2];
    const float* bl    = (const float*)d_in[3];
    const float* Wr    = (const float*)d_in[4];
    const float* gamma = (const float*)d_in[5];
    const float* beta  = (const float*)d_in[6];
    float* out = (float*)d_out;

    const int N = in_sizes[0] / F_DIM;
    const int E = in_sizes[1] / 2;

    // workspace layout: agg[N*128] | deg[N] | colsum[128] | colsumsq[128] | Wt[32768]
    float* agg      = (float*)d_ws;
    float* deg      = agg + (long)N * F_DIM;
    float* colsum   = deg + N;
    float* colsumsq = colsum + H_DIM;
    float2* Wt      = (float2*)(colsumsq + H_DIM);

    long ztotal = (long)N * F_DIM + N + 2 * H_DIM;
    k_zero<<<(int)((ztotal + 255) / 256), 256, 0, stream>>>((float*)d_ws, ztotal);

    k_pack<<<(64 * 2 * 128) / 256, 256, 0, stream>>>(Wl, Wr, Wt);

    long sthreads = (long)E * 32;
    k_scatter<<<(int)((sthreads + 255) / 256), 256, 0, stream>>>(x, ei, agg, deg, E);

    k_gemm<<<N / 16, 256, 0, stream>>>(x, agg, deg, (const v2f*)Wt, bl, out,
                                       colsum, colsumsq);

    long bthreads = (long)N * (H_DIM / 4);
    k_bn<<<(int)((bthreads + 255) / 256), 256, 0, stream>>>(out, colsum, colsumsq,
                                                            gamma, beta, N);
}